// PaiNN_50208167690285
// MI455X (gfx1250) — compile-verified
//
#include <hip/hip_runtime.h>
#include <hip/hip_bf16.h>
#include <math.h>

// ---------------------------------------------------------------------------
// PaiNN message+update step for MI455X (gfx1250), full f32, WMMA f32 16x16x4.
// ---------------------------------------------------------------------------

typedef __attribute__((ext_vector_type(2))) float v2f;
typedef __attribute__((ext_vector_type(8))) float v8f;

static constexpr int NN = 25000;      // nodes
static constexpr int NE = 400000;     // edges
static constexpr int HH = 128;        // feature dim

__device__ __forceinline__ float clip100(float x) {
    return fminf(fmaxf(x, -100.0f), 100.0f);
}

// ---------------------------------------------------------------------------
// Generic C(M,N) = act(A(M,K) @ W(K,N) + bias(N)); row-major everywhere.
// One 16x16 output tile per wave; 8 waves/block cover 128 columns.
// Requires: K in {128, 256} (power of two), N % 128 == 0. M arbitrary.
//
// The 16xK A tile is shared by all 8 waves of the block: it is copied once
// into LDS with a fully-coalesced contiguous load (the tile is 16 consecutive
// rows of A), with row stride K+4 words so the WMMA fragment reads
// (banks = 4*lm + k + 2*half mod 64) are conflict-free.
//
// Fragment layouts (CDNA5 ISA 7.12.2, wave32):
//   A 16x4 : lane L -> row M=L&15, VGPR{0,1} = K = {0,1}+2*(L>>4)
//   B 4x16 : lane L -> col N=L&15, VGPR{0,1} = K = {0,1}+2*(L>>4)
//   C 16x16: lane L -> col N=L&15, VGPR r    = row M = r + 8*(L>>4)
// ---------------------------------------------------------------------------
template <bool SILU, bool BIAS>
__global__ void gemm_wmma_f32(const float* __restrict__ A,
                              const float* __restrict__ W,
                              const float* __restrict__ bias,
                              float* __restrict__ C,
                              int M, int K, int N)
{
    __shared__ float As[16 * 260];   // 16 rows, stride K+4 words (max K=256)

    const int tid  = threadIdx.x;
    const int wave = tid >> 5;
    const int lane = tid & 31;
    const int half = lane >> 4;      // 0: lanes 0-15, 1: lanes 16-31
    const int lm   = lane & 15;

    const int m0  = blockIdx.x * 16;
    const int n0  = (blockIdx.y * 8 + wave) * 16;
    const int lds = K + 4;

    // Cooperative, fully-coalesced copy of the 16xK A tile into LDS.
    // Rows >= M read clamped data; their outputs are discarded at the store.
    {
        const size_t base  = (size_t)m0 * K;
        const size_t amax  = (size_t)M * K - 1;
        const int kmask    = K - 1;
        const int kshift   = (K == 128) ? 7 : 8;
        const int tile     = 16 * K;
        for (int i = tid; i < tile; i += 256) {
            size_t gi = base + (size_t)i;
            if (gi > amax) gi = amax;
            As[(i >> kshift) * lds + (i & kmask)] = A[gi];
        }
    }
    __syncthreads();

    const float* ap = &As[lm * lds + 2 * half];              // step +4 / iter
    const float* bp = W + (size_t)(2 * half) * N + n0 + lm;  // step +4N / iter
    const size_t bstep = (size_t)4 * N;

    v8f acc = {};
#pragma unroll 4
    for (int k = 0; k < K; k += 4) {
        float2 av = *reinterpret_cast<const float2*>(ap);    // ds_load_b64
        v2f a; a.x = av.x; a.y = av.y;
        v2f b;
        b.x = bp[0];
        b.y = bp[(size_t)N];
        // 8 args: (neg_a, A, neg_b, B, c_mod, C, reuse_a, reuse_b)
        acc = __builtin_amdgcn_wmma_f32_16x16x4_f32(
                  false, a, false, b, (short)0, acc, false, false);
        ap += 4;
        bp += bstep;
    }

    const float bval = BIAS ? bias[n0 + lm] : 0.0f;
    float* cp = C + (size_t)(m0 + 8 * half) * N + n0 + lm;
#pragma unroll
    for (int r = 0; r < 8; ++r) {
        const int m = m0 + 8 * half + r;
        if (m < M) {
            float y = acc[r] + bval;
            if (SILU) y = y * (1.0f / (1.0f + __expf(-y)));
            cp[(size_t)r * N] = y;
        }
    }
}

// ---------------------------------------------------------------------------
// Edge message kernel: one wave per edge, lane owns 4 channels (float4).
//   ds_e = Wij[0:128]   * x[r][0:128]
//   dv_e = Wij[128:256] * x[r][128:256] * dir[a]
//        + Wij[256:384] * x[r][256:384] * v[r][a]
// scatter-add into ds_acc[snd], dv_acc[snd] with global_atomic_add_f32.
// ---------------------------------------------------------------------------
__global__ void edge_kernel(const float* __restrict__ x,      // (NN,384)
                            const float* __restrict__ v,      // (NN,3,128)
                            const float* __restrict__ Wij,    // (NE,384)
                            const float* __restrict__ dir_ij, // (NE,3)
                            const int*   __restrict__ senders,
                            const int*   __restrict__ receivers,
                            float* __restrict__ ds_acc,       // (NN,128)
                            float* __restrict__ dv_acc,       // (NN,3,128)
                            int n_edges)
{
    const int e = (int)((blockIdx.x * (size_t)blockDim.x + threadIdx.x) >> 5);
    if (e >= n_edges) return;                 // wave-uniform
    const int lane = threadIdx.x & 31;
    const int c = lane * 4;

    const int r   = receivers[e];
    const int snd = senders[e];

    const float4* wp = reinterpret_cast<const float4*>(Wij + (size_t)e * 384);
    const float4* xp = reinterpret_cast<const float4*>(x + (size_t)r * 384);
    const float4 w0 = wp[lane], w1 = wp[lane + 32], w2 = wp[lane + 64];
    const float4 x0 = xp[lane], x1 = xp[lane + 32], x2 = xp[lane + 64];

    const float d0 = dir_ij[e * 3 + 0];
    const float d1 = dir_ij[e * 3 + 1];
    const float d2 = dir_ij[e * 3 + 2];

    // ds_e
    float* dsp = ds_acc + (size_t)snd * HH + c;
    atomicAdd(dsp + 0, w0.x * x0.x);
    atomicAdd(dsp + 1, w0.y * x0.y);
    atomicAdd(dsp + 2, w0.z * x0.z);
    atomicAdd(dsp + 3, w0.w * x0.w);

    // gates
    const float g1x = w1.x * x1.x, g1y = w1.y * x1.y, g1z = w1.z * x1.z, g1w = w1.w * x1.w;
    const float g2x = w2.x * x2.x, g2y = w2.y * x2.y, g2z = w2.z * x2.z, g2w = w2.w * x2.w;

    const float dd[3] = {d0, d1, d2};
#pragma unroll
    for (int a = 0; a < 3; ++a) {
        const float4 vj =
            reinterpret_cast<const float4*>(v + ((size_t)r * 3 + a) * HH)[lane];
        float* dvp = dv_acc + ((size_t)snd * 3 + a) * HH + c;
        atomicAdd(dvp + 0, g1x * dd[a] + g2x * vj.x);
        atomicAdd(dvp + 1, g1y * dd[a] + g2y * vj.y);
        atomicAdd(dvp + 2, g1z * dd[a] + g2z * vj.z);
        atomicAdd(dvp + 3, g1w * dd[a] + g2w * vj.w);
    }
}

// ---------------------------------------------------------------------------
// Elementwise helpers
// ---------------------------------------------------------------------------
__global__ void fill_zero(float* __restrict__ p, size_t n)
{
    size_t i = blockIdx.x * (size_t)blockDim.x + threadIdx.x;
    const size_t stride = (size_t)gridDim.x * blockDim.x;
    for (; i < n; i += stride) p[i] = 0.0f;
}

// out = base + clip(delta, +-100)
__global__ void add_clip_kernel(const float* __restrict__ base,
                                const float* __restrict__ delta,
                                float* __restrict__ out, size_t n)
{
    size_t i = blockIdx.x * (size_t)blockDim.x + threadIdx.x;
    if (i < n) out[i] = base[i] + clip100(delta[i]);
}

// ts = concat(s1, ||v_r||) ; vw = (NN,3,256) = [v_l | v_r]
__global__ void norm_concat_kernel(const float* __restrict__ s1,  // (NN,128)
                                   const float* __restrict__ vw,  // (NN,3,256)
                                   float* __restrict__ ts,        // (NN,256)
                                   int n_nodes)
{
    const int i = blockIdx.x * blockDim.x + threadIdx.x;   // node*128 + c
    if (i >= n_nodes * HH) return;
    const int node = i >> 7;
    const int c    = i & 127;
    const float* vr = vw + (size_t)node * 3 * 256 + 128 + c;
    const float a0 = vr[0], a1 = vr[256], a2 = vr[512];
    const float nrm = sqrtf(a0 * a0 + a1 * a1 + a2 * a2 + 1e-8f);
    ts[(size_t)node * 256 + c] = s1[i];
    ts[(size_t)node * 256 + 128 + c] = nrm;
}

// Final epilogue (in-place on d_out which holds s1, v1):
//   dot   = sum_a v_l[a]*v_r[a]
//   s    += clip(ds_u + dsv*dot)
//   v[a] += clip(v_l[a]*gate)
__global__ void final_kernel(const float* __restrict__ a3,   // (NN,384)
                             const float* __restrict__ vw,   // (NN,3,256)
                             float* __restrict__ s_io,       // (NN,128)
                             float* __restrict__ v_io,       // (NN,3,128)
                             int n_nodes)
{
    const int i = blockIdx.x * blockDim.x + threadIdx.x;
    if (i >= n_nodes * HH) return;
    const int node = i >> 7;
    const int c    = i & 127;

    const float* vwp = vw + (size_t)node * 3 * 256;
    const float vl0 = vwp[c],       vr0 = vwp[128 + c];
    const float vl1 = vwp[256 + c], vr1 = vwp[384 + c];
    const float vl2 = vwp[512 + c], vr2 = vwp[640 + c];
    const float dot = vl0 * vr0 + vl1 * vr1 + vl2 * vr2;

    const float* ap = a3 + (size_t)node * 384;
    const float ds_u = ap[c];
    const float gate = ap[128 + c];
    const float dsv  = ap[256 + c] * dot;

    s_io[i] += clip100(ds_u + dsv);

    float* vp = v_io + (size_t)node * 3 * HH + c;
    vp[0]   += clip100(vl0 * gate);
    vp[128] += clip100(vl1 * gate);
    vp[256] += clip100(vl2 * gate);
}

// ---------------------------------------------------------------------------
template <bool SILU, bool BIAS>
static void launch_gemm(const float* A, const float* W, const float* bias,
                        float* C, int M, int K, int N, hipStream_t stream)
{
    dim3 grid((M + 15) / 16, N / 128);
    gemm_wmma_f32<SILU, BIAS><<<grid, 256, 0, stream>>>(A, W, bias, C, M, K, N);
}

extern "C" void kernel_launch(void* const* d_in, const int* in_sizes, int n_in,
                              void* d_out, int out_size, void* d_ws, size_t ws_size,
                              hipStream_t stream)
{
    (void)in_sizes; (void)n_in; (void)out_size; (void)ws_size;

    const float* s      = (const float*)d_in[0];   // (NN,1,128)
    const float* v      = (const float*)d_in[1];   // (NN,3,128)
    const float* dir_ij = (const float*)d_in[2];   // (NE,3)
    const float* Wij    = (const float*)d_in[3];   // (NE,1,384)
    const float* Wi1    = (const float*)d_in[4];   // (128,128)
    const float* bi1    = (const float*)d_in[5];   // (128,)
    const float* Wi2    = (const float*)d_in[6];   // (128,384)
    const float* bi2    = (const float*)d_in[7];   // (384,)
    const float* Wm1    = (const float*)d_in[8];   // (256,128)
    const float* bm1    = (const float*)d_in[9];   // (128,)
    const float* Wm2    = (const float*)d_in[10];  // (128,384)
    const float* bm2    = (const float*)d_in[11];  // (384,)
    const float* Wv     = (const float*)d_in[12];  // (128,256)
    const int*   snd    = (const int*)d_in[13];    // (NE,)
    const int*   rcv    = (const int*)d_in[14];    // (NE,)

    float* outS = (float*)d_out;                   // (NN,128)
    float* outV = outS + (size_t)NN * HH;          // (NN,3,128)

    // Workspace layout (lifetime-based reuse, total 128 MB):
    //   bufA [76.8MB]: x (38.4) | dv (38.4)  -> later vw (NN,3,256)
    //   bufB [12.8MB]: h -> ds -> hid
    //   bufC [38.4MB]: ts (25.6) -> a3 (38.4)
    float* bufA = (float*)d_ws;
    float* bufB = (float*)((char*)d_ws + 76800000);
    float* bufC = (float*)((char*)d_ws + 89600000);

    float* x   = bufA;                          // (NN,384)
    float* dv  = bufA + (size_t)NN * 384;       // (NN,3,128)
    float* h1  = bufB;                          // (NN,128)
    float* ds  = bufB;                          // (NN,128)
    float* hid = bufB;                          // (NN,128)
    float* vw  = bufA;                          // (NN,3,256)
    float* ts  = bufC;                          // (NN,256)
    float* a3  = bufC;                          // (NN,384)

    const size_t nS = (size_t)NN * HH;          // 3.2M
    const size_t nV = (size_t)NN * 3 * HH;      // 9.6M

    // ---- message phase ----
    // 1) h = silu(s @ Wi1 + bi1)
    launch_gemm<true, true>(s, Wi1, bi1, h1, NN, 128, 128, stream);
    // 2) x = h @ Wi2 + bi2
    launch_gemm<false, true>(h1, Wi2, bi2, x, NN, 128, 384, stream);
    // 3) zero accumulators
    fill_zero<<<2048, 256, 0, stream>>>(ds, nS);
    fill_zero<<<2048, 256, 0, stream>>>(dv, nV);
    // 4) edge gather/combine/scatter (1 wave per edge, 8 edges per block)
    edge_kernel<<<(NE + 7) / 8, 256, 0, stream>>>(x, v, Wij, dir_ij, snd, rcv,
                                                  ds, dv, NE);
    // 5) s1 = s + clip(ds); v1 = v + clip(dv)  (into d_out)
    add_clip_kernel<<<(int)((nS + 255) / 256), 256, 0, stream>>>(s, ds, outS, nS);
    add_clip_kernel<<<(int)((nV + 255) / 256), 256, 0, stream>>>(v, dv, outV, nV);

    // ---- update phase ----
    // 6) vw = v1 @ Wv      (treat v1 as (3*NN, 128))
    launch_gemm<false, false>(outV, Wv, nullptr, vw, 3 * NN, 128, 256, stream);
    // 7) ts = concat(s1, ||v_r||)
    norm_concat_kernel<<<(NN * HH + 255) / 256, 256, 0, stream>>>(outS, vw, ts, NN);
    // 8) hid = silu(ts @ Wm1 + bm1)
    launch_gemm<true, true>(ts, Wm1, bm1, hid, NN, 256, 128, stream);
    // 9) a3 = hid @ Wm2 + bm2
    launch_gemm<false, true>(hid, Wm2, bm2, a3, NN, 128, 384, stream);
    // 10) final epilogue (in-place on d_out)
    final_kernel<<<(NN * HH + 255) / 256, 256, 0, stream>>>(a3, vw, outS, outV, NN);
}